// DualLoss_55130200211587
// MI455X (gfx1250) — compile-verified
//
#include <hip/hip_runtime.h>
#include <hip/hip_bf16.h>
#include <math.h>

#define NUM_CLASSES 1000
#define DENSE_DIM   4096
#define BATCH       8192

#define TILE_D      32                    // dims per block tile
#define NTILES      (DENSE_DIM / TILE_D)  // 128
#define SCHUNKS     4                     // batch splits
#define CHUNK       (BATCH / SCHUNKS)     // 2048
#define MBLOCK      256

typedef __attribute__((ext_vector_type(2))) float v2f;
typedef __attribute__((ext_vector_type(8))) float v8f;

// Full-wave32 sum reduction using V_WMMA_F32_16X16X4_F32.
// A: vgpr0 = lane partial, vgpr1 = 0  => A[m,0]=x[m], A[m,2]=x[m+16], A[m,{1,3}]=0
// B: all ones  => D[m,n] = x[m] + x[m+16]
// Lane L<16 holds D rows 0..7 (col L), lane L+16 holds rows 8..15.
// sum of 8 D regs + shfl_xor(16) == sum over all 32 lanes, broadcast to every lane.
// Requires EXEC == all ones (call only in fully-converged code).
__device__ __forceinline__ float wave_sum32(float x) {
    v2f a; a[0] = x;    a[1] = 0.0f;
    v2f b; b[0] = 1.0f; b[1] = 1.0f;
    v8f c = {};
    c = __builtin_amdgcn_wmma_f32_16x16x4_f32(false, a, false, b,
                                              (short)0, c, false, false);
    float t = c[0] + c[1] + c[2] + c[3] + c[4] + c[5] + c[6] + c[7];
    t += __shfl_xor(t, 16, 32);
    return t;
}

// One pass over dense_out: sigmoid -> LDS class-tile segment sum (ds_add_f32),
// fused BCE-with-logits partial sum (WMMA wave reduce), then atomic flush.
__global__ __launch_bounds__(MBLOCK)
void fused_dense_kernel(const float* __restrict__ dense_out,
                        const int*   __restrict__ target,
                        const float* __restrict__ labels,
                        float* __restrict__ out_dense,   // d_out + 1, [C, D]
                        float* __restrict__ ws)          // ws[1] += bce partial
{
    __shared__ float acc[NUM_CLASSES * TILE_D];          // 125 KB (CDNA5 big LDS)

    const int tid   = threadIdx.x;
    const int dtile = blockIdx.x & (NTILES - 1);
    const int chunk = blockIdx.x >> 7;                   // blockIdx / NTILES
    const int d0    = dtile * TILE_D;
    const int s0    = chunk * CHUNK;

    for (int i = tid; i < NUM_CLASSES * TILE_D; i += MBLOCK) acc[i] = 0.0f;
    __syncthreads();

    float bce = 0.0f;
    for (int i = tid; i < CHUNK * TILE_D; i += MBLOCK) {
        const int s   = s0 + (i >> 5);                   // sample (wave-uniform)
        const int d   = i & (TILE_D - 1);                // lane-consecutive dim
        const int col = d0 + d;

        const float x = dense_out[(size_t)s * DENSE_DIM + col];
        // prefetch next iteration's element (8 samples ahead); clamp to stay in-buffer
        const int sp = (s + (MBLOCK / TILE_D) < BATCH) ? (s + (MBLOCK / TILE_D)) : s;
        __builtin_prefetch(&dense_out[(size_t)sp * DENSE_DIM + col], 0, 1);

        const int   cls = target[s];
        const float y   = labels[(size_t)cls * DENSE_DIM + col];

        const float sig = 1.0f / (1.0f + expf(-x));
        atomicAdd(&acc[cls * TILE_D + d], sig);          // ds_add_f32, bank-conflict-free

        bce += fmaxf(x, 0.0f) - x * y + log1pf(expf(-fabsf(x)));
    }

    // fully converged here: WMMA wave reduction of BCE partials
    const float wsum = wave_sum32(bce);
    if ((tid & 31) == 0) atomicAdd(&ws[1], wsum);

    __syncthreads();
    for (int i = tid; i < NUM_CLASSES * TILE_D; i += MBLOCK) {
        const int c = i >> 5;
        const int d = i & (TILE_D - 1);
        atomicAdd(&out_dense[(size_t)c * DENSE_DIM + d0 + d], acc[i]);
    }
}

// One wave per row: log-softmax CE + class histogram.
__global__ __launch_bounds__(256)
void ce_kernel(const float* __restrict__ logits,
               const int*   __restrict__ target,
               float* __restrict__ total_acc,            // d_out + 1 + C*D, [C]
               float* __restrict__ ws)                   // ws[0] += ce partial
{
    __shared__ float part[8];
    const int tid  = threadIdx.x;
    const int lane = tid & 31;
    const int row  = blockIdx.x * 8 + (tid >> 5);
    const float* lr = logits + (size_t)row * NUM_CLASSES;

    float m = -INFINITY;
    for (int j = lane; j < NUM_CLASSES; j += 32) m = fmaxf(m, lr[j]);
#pragma unroll
    for (int off = 16; off > 0; off >>= 1) m = fmaxf(m, __shfl_xor(m, off, 32));

    float ssum = 0.0f;
    for (int j = lane; j < NUM_CLASSES; j += 32) ssum += expf(lr[j] - m);
    ssum = wave_sum32(ssum);                             // converged after loop

    if (lane == 0) {
        const int t = target[row];
        part[tid >> 5] = (m + logf(ssum)) - lr[t];
        atomicAdd(&total_acc[t], 1.0f);
    }
    __syncthreads();

    if (tid < 32) {                                      // wave 0 fully active
        float v = (lane < 8) ? part[lane] : 0.0f;
        v = wave_sum32(v);
        if (lane == 0) atomicAdd(&ws[0], v);
    }
}

__global__ void finalize_kernel(const float* __restrict__ ws,
                                float* __restrict__ out)
{
    out[0] = 0.5f * (ws[0] / (float)BATCH)
           + 0.5f * (ws[1] / ((float)BATCH * (float)DENSE_DIM));
}

extern "C" void kernel_launch(void* const* d_in, const int* in_sizes, int n_in,
                              void* d_out, int out_size, void* d_ws, size_t ws_size,
                              hipStream_t stream) {
    const float* logits    = (const float*)d_in[0];  // [8192, 1000]
    const float* dense_out = (const float*)d_in[1];  // [8192, 4096]
    const int*   target    = (const int*)  d_in[2];  // [8192]
    const float* labels    = (const float*)d_in[3];  // [1000, 4096]

    float* out = (float*)d_out;                      // [1 + C*D + C]
    float* ws  = (float*)d_ws;                       // ws[0]=ce sum, ws[1]=bce sum

    hipMemsetAsync(d_out, 0, (size_t)out_size * sizeof(float), stream);
    hipMemsetAsync(d_ws, 0, 2 * sizeof(float), stream);

    fused_dense_kernel<<<NTILES * SCHUNKS, MBLOCK, 0, stream>>>(
        dense_out, target, labels, out + 1, ws);

    ce_kernel<<<BATCH / 8, 256, 0, stream>>>(
        logits, target, out + 1 + (size_t)NUM_CLASSES * DENSE_DIM, ws);

    finalize_kernel<<<1, 1, 0, stream>>>(ws, out);
}